// FedGSL_gin_69320772157914
// MI455X (gfx1250) — compile-verified
//
#include <hip/hip_runtime.h>
#include <hip/hip_bf16.h>

// ---------------------------------------------------------------------------
// GIN (2x GINConv + BN(batch stats) + ReLU) + concat + final Linear
// N=100000 nodes, E=1600000 edges, all dims 64 (final in = 128).
// f32 WMMA (v_wmma_f32_16x16x4_f32) for every GEMM; f32 atomics for the
// scatter-add; LDS-staged weights; fused BN+ReLU into the 2nd GEMM's A-frag.
// ---------------------------------------------------------------------------

typedef float v2f __attribute__((ext_vector_type(2)));
typedef float v8f __attribute__((ext_vector_type(8)));

#define DIM 64
#define BN_EPS 1e-5f

__device__ __forceinline__ v8f wmma4(v2f a, v2f b, v8f c) {
    // D(16x16) = A(16x4) x B(4x16) + C   -- exact f32 math
    return __builtin_amdgcn_wmma_f32_16x16x4_f32(
        /*neg_a=*/false, a, /*neg_b=*/false, b,
        /*c_mod=*/(short)0, c, /*reuse_a=*/false, /*reuse_b=*/false);
}

// ---------------------------------------------------------------- zero fill
__global__ void zero_kernel(float* __restrict__ p, long long count) {
    long long i = (long long)blockIdx.x * blockDim.x + threadIdx.x;
    long long stride = (long long)gridDim.x * blockDim.x;
    for (; i < count; i += stride) p[i] = 0.0f;
}

// ------------------------------------------------- edge scatter: agg[dst]+=x[src]
// One work item = (edge, 4 channels). float4 gather + 4 f32 atomics.
__global__ void scatter_kernel(const float* __restrict__ x,
                               const int* __restrict__ src,
                               const int* __restrict__ dst,
                               float* __restrict__ agg, int nedges) {
    long long total = (long long)nedges * 16;
    long long stride = (long long)gridDim.x * blockDim.x;
    for (long long idx = (long long)blockIdx.x * blockDim.x + threadIdx.x;
         idx < total; idx += stride) {
        int e  = (int)(idx >> 4);
        int c4 = ((int)idx & 15) << 2;
        int s = src[e];
        int d = dst[e];
        float4 v = *(const float4*)(x + (size_t)s * DIM + c4);
        float* p = agg + (size_t)d * DIM + c4;
        atomicAdd(p + 0, v.x);
        atomicAdd(p + 1, v.y);
        atomicAdd(p + 2, v.z);
        atomicAdd(p + 3, v.w);
    }
}

// ---------------------------------------------- GEMM1: out = (agg + x) @ W + b
// One wave per 16-row tile; full 64-wide output in 4 v8f accumulators.
__global__ void gin_lin1_kernel(const float* __restrict__ agg,
                                const float* __restrict__ x,
                                const float* __restrict__ W,   // [64,64] row-major (in,out)
                                const float* __restrict__ b,   // [64]
                                float* __restrict__ out, int ntiles) {
    __shared__ float sW[DIM * DIM];
    int tid = threadIdx.x;
#pragma unroll
    for (int i = 0; i < 4; ++i)
        ((float4*)sW)[tid + i * 256] = ((const float4*)W)[tid + i * 256];
    __syncthreads();

    int wave = tid >> 5;
    int lane = tid & 31;
    int tile = blockIdx.x * 8 + wave;
    if (tile >= ntiles) return;

    int m  = lane & 15;   // row within tile (A), col within n-tile (B/D)
    int kh = lane >> 4;   // K-half selector
    size_t rowoff = ((size_t)tile * 16 + m) * DIM;

    v8f acc[4];
#pragma unroll
    for (int n = 0; n < 4; ++n) {
        float bb = b[n * 16 + m];
        acc[n] = (v8f){bb, bb, bb, bb, bb, bb, bb, bb};
    }

#pragma unroll
    for (int k = 0; k < 16; ++k) {
        int kc = k * 4 + kh * 2;
        float2 a1 = *(const float2*)(agg + rowoff + kc);
        float2 a2 = *(const float2*)(x + rowoff + kc);
        v2f a; a.x = a1.x + a2.x; a.y = a1.y + a2.y;
#pragma unroll
        for (int n = 0; n < 4; ++n) {
            v2f bf;
            bf.x = sW[kc * DIM + n * 16 + m];
            bf.y = sW[(kc + 1) * DIM + n * 16 + m];
            acc[n] = wmma4(a, bf, acc[n]);
        }
    }

    size_t outbase = ((size_t)tile * 16 + kh * 8) * DIM + m;
#pragma unroll
    for (int n = 0; n < 4; ++n)
#pragma unroll
        for (int v = 0; v < 8; ++v)
            out[outbase + (size_t)v * DIM + n * 16] = acc[n][v];
}

// ------------------------------------- BN stats: per-channel sum & sum of squares
#define STAT_ROWS 256
__global__ void bn_stats_kernel(const float* __restrict__ h,
                                float* __restrict__ gstats, int nrows) {
    __shared__ float sred[4][2][DIM];
    int tid = threadIdx.x;
    int c = tid & 63;
    int g = tid >> 6;  // 0..3
    int base = blockIdx.x * STAT_ROWS;
    int rend = base + STAT_ROWS; if (rend > nrows) rend = nrows;
    float s = 0.0f, ss = 0.0f;
    for (int r = base + g; r < rend; r += 4) {
        float v = h[(size_t)r * DIM + c];
        s += v; ss += v * v;
    }
    sred[g][0][c] = s;
    sred[g][1][c] = ss;
    __syncthreads();
    if (tid < 64) {
        float t = sred[0][0][tid] + sred[1][0][tid] + sred[2][0][tid] + sred[3][0][tid];
        atomicAdd(&gstats[tid], t);
    } else if (tid < 128) {
        int cc = tid - 64;
        float t = sred[0][1][cc] + sred[1][1][cc] + sred[2][1][cc] + sred[3][1][cc];
        atomicAdd(&gstats[DIM + cc], t);
    }
}

// gstats[c] -> scale, gstats[64+c] -> shift
__global__ void bn_finalize_kernel(float* __restrict__ gstats,
                                   const float* __restrict__ gamma,
                                   const float* __restrict__ beta, int nrows) {
    int c = threadIdx.x;
    if (c < DIM) {
        float inv_n = 1.0f / (float)nrows;
        float mu  = gstats[c] * inv_n;
        float var = gstats[DIM + c] * inv_n - mu * mu;
        float sc  = gamma[c] * rsqrtf(var + BN_EPS);
        gstats[c] = sc;
        gstats[DIM + c] = beta[c] - mu * sc;
    }
}

// --------------------- GEMM2: out = relu?(relu(h*scale+shift) @ W + b)
__global__ void gin_lin2_kernel(const float* __restrict__ h,
                                const float* __restrict__ gstats,  // [scale|shift]
                                const float* __restrict__ W,
                                const float* __restrict__ b,
                                float* __restrict__ out, int ntiles, int relu_out) {
    __shared__ float sW[DIM * DIM];
    __shared__ float sS[2 * DIM];
    int tid = threadIdx.x;
#pragma unroll
    for (int i = 0; i < 4; ++i)
        ((float4*)sW)[tid + i * 256] = ((const float4*)W)[tid + i * 256];
    if (tid < 128) sS[tid] = gstats[tid];
    __syncthreads();

    int wave = tid >> 5;
    int lane = tid & 31;
    int tile = blockIdx.x * 8 + wave;
    if (tile >= ntiles) return;

    int m  = lane & 15;
    int kh = lane >> 4;
    size_t rowoff = ((size_t)tile * 16 + m) * DIM;

    v8f acc[4];
#pragma unroll
    for (int n = 0; n < 4; ++n) {
        float bb = b[n * 16 + m];
        acc[n] = (v8f){bb, bb, bb, bb, bb, bb, bb, bb};
    }

#pragma unroll
    for (int k = 0; k < 16; ++k) {
        int kc = k * 4 + kh * 2;
        float2 hv = *(const float2*)(h + rowoff + kc);
        v2f a;
        a.x = fmaxf(hv.x * sS[kc]     + sS[DIM + kc],     0.0f);
        a.y = fmaxf(hv.y * sS[kc + 1] + sS[DIM + kc + 1], 0.0f);
#pragma unroll
        for (int n = 0; n < 4; ++n) {
            v2f bf;
            bf.x = sW[kc * DIM + n * 16 + m];
            bf.y = sW[(kc + 1) * DIM + n * 16 + m];
            acc[n] = wmma4(a, bf, acc[n]);
        }
    }

    size_t outbase = ((size_t)tile * 16 + kh * 8) * DIM + m;
#pragma unroll
    for (int n = 0; n < 4; ++n)
#pragma unroll
        for (int v = 0; v < 8; ++v) {
            float val = acc[n][v];
            if (relu_out) val = fmaxf(val, 0.0f);
            out[outbase + (size_t)v * DIM + n * 16] = val;
        }
}

// ------------- final: out = [xl | glob] (N x 128) @ lin_W (128x64) + lin_b
__global__ void final_lin_kernel(const float* __restrict__ xl,
                                 const float* __restrict__ glob,
                                 const float* __restrict__ W,   // [128,64]
                                 const float* __restrict__ b,   // [64]
                                 float* __restrict__ out, int ntiles) {
    __shared__ float sW[128 * DIM];
    int tid = threadIdx.x;
#pragma unroll
    for (int i = 0; i < 8; ++i)
        ((float4*)sW)[tid + i * 256] = ((const float4*)W)[tid + i * 256];
    __syncthreads();

    int wave = tid >> 5;
    int lane = tid & 31;
    int tile = blockIdx.x * 8 + wave;
    if (tile >= ntiles) return;

    int m  = lane & 15;
    int kh = lane >> 4;
    size_t rowoff = ((size_t)tile * 16 + m) * DIM;

    v8f acc[4];
#pragma unroll
    for (int n = 0; n < 4; ++n) {
        float bb = b[n * 16 + m];
        acc[n] = (v8f){bb, bb, bb, bb, bb, bb, bb, bb};
    }

    // K = 0..63 from xl
#pragma unroll
    for (int k = 0; k < 16; ++k) {
        int kc = k * 4 + kh * 2;
        float2 av = *(const float2*)(xl + rowoff + kc);
        v2f a; a.x = av.x; a.y = av.y;
#pragma unroll
        for (int n = 0; n < 4; ++n) {
            v2f bf;
            bf.x = sW[kc * DIM + n * 16 + m];
            bf.y = sW[(kc + 1) * DIM + n * 16 + m];
            acc[n] = wmma4(a, bf, acc[n]);
        }
    }
    // K = 64..127 from glob
#pragma unroll
    for (int k = 0; k < 16; ++k) {
        int kc = k * 4 + kh * 2;
        float2 av = *(const float2*)(glob + rowoff + kc);
        v2f a; a.x = av.x; a.y = av.y;
#pragma unroll
        for (int n = 0; n < 4; ++n) {
            v2f bf;
            bf.x = sW[(DIM + kc) * DIM + n * 16 + m];
            bf.y = sW[(DIM + kc + 1) * DIM + n * 16 + m];
            acc[n] = wmma4(a, bf, acc[n]);
        }
    }

    size_t outbase = ((size_t)tile * 16 + kh * 8) * DIM + m;
#pragma unroll
    for (int n = 0; n < 4; ++n)
#pragma unroll
        for (int v = 0; v < 8; ++v)
            out[outbase + (size_t)v * DIM + n * 16] = acc[n][v];
}

// ---------------------------------------------------------------------------
extern "C" void kernel_launch(void* const* d_in, const int* in_sizes, int n_in,
                              void* d_out, int out_size, void* d_ws, size_t ws_size,
                              hipStream_t stream) {
    const float* batch_x = (const float*)d_in[0];
    const int*   eidx    = (const int*)d_in[1];   // [2, E]
    const float* glob    = (const float*)d_in[2];
    const float* W1_0 = (const float*)d_in[3];
    const float* b1_0 = (const float*)d_in[4];
    const float* g_0  = (const float*)d_in[5];
    const float* be_0 = (const float*)d_in[6];
    const float* W2_0 = (const float*)d_in[7];
    const float* b2_0 = (const float*)d_in[8];
    const float* W1_1 = (const float*)d_in[9];
    const float* b1_1 = (const float*)d_in[10];
    const float* g_1  = (const float*)d_in[11];
    const float* be_1 = (const float*)d_in[12];
    const float* W2_1 = (const float*)d_in[13];
    const float* b2_1 = (const float*)d_in[14];
    const float* lin_W = (const float*)d_in[15];
    const float* lin_b = (const float*)d_in[16];
    float* out = (float*)d_out;

    const int n = in_sizes[0] / DIM;       // 100000
    const int e = in_sizes[1] / 2;         // 1600000
    const int ntiles = n / 16;             // 6250 (n % 16 == 0)
    const int*  src = eidx;
    const int*  dst = eidx + e;

    float* A     = (float*)d_ws;           // agg / layer1-out  [n*64]
    float* B     = A + (size_t)n * DIM;    // pre-BN hidden     [n*64]
    float* C     = B + (size_t)n * DIM;    // layer0-out        [n*64]
    float* stats = C + (size_t)n * DIM;    // [128]

    dim3 blk(256);
    dim3 gemm_grid((ntiles + 7) / 8);
    dim3 stat_grid((n + STAT_ROWS - 1) / STAT_ROWS);
    dim3 zero_grid(((long long)n * DIM + 255) / 256);
    dim3 scat_grid(8192);

    // -------- layer 0 --------
    zero_kernel<<<zero_grid, blk, 0, stream>>>(A, (long long)n * DIM);
    zero_kernel<<<1, 128, 0, stream>>>(stats, 128);
    scatter_kernel<<<scat_grid, blk, 0, stream>>>(batch_x, src, dst, A, e);
    gin_lin1_kernel<<<gemm_grid, blk, 0, stream>>>(A, batch_x, W1_0, b1_0, B, ntiles);
    bn_stats_kernel<<<stat_grid, blk, 0, stream>>>(B, stats, n);
    bn_finalize_kernel<<<1, 64, 0, stream>>>(stats, g_0, be_0, n);
    gin_lin2_kernel<<<gemm_grid, blk, 0, stream>>>(B, stats, W2_0, b2_0, C, ntiles,
                                                   /*relu_out=*/1);

    // -------- layer 1 --------
    zero_kernel<<<zero_grid, blk, 0, stream>>>(A, (long long)n * DIM);
    zero_kernel<<<1, 128, 0, stream>>>(stats, 128);
    scatter_kernel<<<scat_grid, blk, 0, stream>>>(C, src, dst, A, e);
    gin_lin1_kernel<<<gemm_grid, blk, 0, stream>>>(A, C, W1_1, b1_1, B, ntiles);
    bn_stats_kernel<<<stat_grid, blk, 0, stream>>>(B, stats, n);
    bn_finalize_kernel<<<1, 64, 0, stream>>>(stats, g_1, be_1, n);
    gin_lin2_kernel<<<gemm_grid, blk, 0, stream>>>(B, stats, W2_1, b2_1, A, ntiles,
                                                   /*relu_out=*/0);

    // -------- final linear on concat([x, glob]) --------
    final_lin_kernel<<<gemm_grid, blk, 0, stream>>>(A, glob, lin_W, lin_b, out, ntiles);
}